// GATConv_6493990552349
// MI455X (gfx1250) — compile-verified
//
#include <hip/hip_runtime.h>
#include <hip/hip_bf16.h>
#include <math.h>

typedef __attribute__((ext_vector_type(2))) float v2f;
typedef __attribute__((ext_vector_type(8))) float v8f;

#define F_IN 128
#define HC   128   // H*C
#define NH   4     // heads
#define NC   32    // channels per head
#define NEG_SLOPE 0.2f

// ---------------------------------------------------------------------------
// Kernel 1: x = nodes @ W + b_lin   via V_WMMA_F32_16X16X4_F32 (fp32 exact)
// One wave per 16x16 output tile. Block = 256 threads = 8 waves = 8 N-tiles.
// A frag (16x4 f32): lane l, reg j -> A[M=l%16, K=kblk + j + 2*(l/16)]
// B frag (4x16 f32): lane l, reg j -> B[K=kblk + j + 2*(l/16), N=l%16]
// D (16x16 f32):     lane l, reg r -> D[M=r + 8*(l/16), N=l%16]
// ---------------------------------------------------------------------------
__global__ void gat_gemm_wmma(const float* __restrict__ A,
                              const float* __restrict__ Wm,
                              const float* __restrict__ blin,
                              float* __restrict__ X, int N) {
    const int lane = threadIdx.x & 31;
    const int wave = threadIdx.x >> 5;          // N-tile 0..7
    const int mBase = blockIdx.x * 16;
    const int nBase = wave * 16;
    const int half = lane >> 4;                 // 0 or 1
    const int lm   = lane & 15;

    int row = mBase + lm;
    if (row >= N) row = N - 1;                  // clamp (N%16==0 here, safety)
    const float* Arow = A + (size_t)row * F_IN;
    const float* Bp   = Wm + nBase + lm;        // column (nBase+lm) of W

    v8f acc = {};
#pragma unroll
    for (int k = 0; k < F_IN; k += 4) {
        const int kk = k + 2 * half;
        v2f a = *reinterpret_cast<const v2f*>(Arow + kk);     // 8B-aligned
        v2f b;
        b.x = Bp[(size_t)(kk + 0) * HC];
        b.y = Bp[(size_t)(kk + 1) * HC];
        acc = __builtin_amdgcn_wmma_f32_16x16x4_f32(
            /*neg_a=*/false, a, /*neg_b=*/false, b,
            /*c_mod=*/(short)0, acc, /*reuse_a=*/false, /*reuse_b=*/false);
    }

    const float bv = blin[nBase + lm];
#pragma unroll
    for (int r = 0; r < 8; ++r) {
        const int orow = mBase + r + 8 * half;
        if (orow < N) X[(size_t)orow * HC + nBase + lm] = acc[r] + bv;
    }
}

// ---------------------------------------------------------------------------
// Kernel 2: per-node attention scores + init of reduction buffers.
// block = 128 threads = one node; wave = head; lane = channel.
// s_src[n,h] = x[n,h,:] . alpha[h,0:32] ; s_dst[n,h] = x[n,h,:] . alpha[h,32:64]
// ---------------------------------------------------------------------------
__global__ void gat_node_scores(const float* __restrict__ X,
                                const float* __restrict__ alpha,
                                float* __restrict__ s_src,
                                float* __restrict__ s_dst,
                                float* __restrict__ seg_max,
                                float* __restrict__ denom,
                                float* __restrict__ out, int N) {
    const int n = blockIdx.x;
    if (n >= N) return;
    const int h = threadIdx.x >> 5;
    const int c = threadIdx.x & 31;

    const float xv = X[(size_t)n * HC + h * NC + c];
    float vs = xv * alpha[h * (2 * NC) + c];
    float vd = xv * alpha[h * (2 * NC) + NC + c];
#pragma unroll
    for (int off = 16; off > 0; off >>= 1) {
        vs += __shfl_xor(vs, off, 32);
        vd += __shfl_xor(vd, off, 32);
    }
    if (c == 0) {
        s_src[n * NH + h]   = vs;
        s_dst[n * NH + h]   = vd;
        seg_max[n * NH + h] = -INFINITY;
        denom[n * NH + h]   = 0.0f;
    }
    out[(size_t)n * HC + threadIdx.x] = 0.0f;   // accumulator init
}

// float atomic max via sign-aware integer atomics (init = -inf)
__device__ __forceinline__ void atomicMaxF(float* addr, float val) {
    if (val >= 0.0f)
        atomicMax(reinterpret_cast<int*>(addr), __float_as_int(val));
    else
        atomicMin(reinterpret_cast<unsigned int*>(addr), __float_as_uint(val));
}

// ---------------------------------------------------------------------------
// Kernel 3: per-edge logits -> atomic max into seg_max[rcv,h].
// Edges E..E+N-1 are the appended self loops.
// ---------------------------------------------------------------------------
__global__ void gat_edge_max(const int* __restrict__ snd,
                             const int* __restrict__ rcv,
                             const float* __restrict__ s_src,
                             const float* __restrict__ s_dst,
                             float* __restrict__ seg_max,
                             int E, int EP) {
    const int e = blockIdx.x * blockDim.x + threadIdx.x;
    if (e >= EP) return;
    const int s = (e < E) ? snd[e] : (e - E);
    const int r = (e < E) ? rcv[e] : (e - E);
#pragma unroll
    for (int h = 0; h < NH; ++h) {
        float lg = s_src[s * NH + h] + s_dst[r * NH + h];
        lg = (lg > 0.0f) ? lg : NEG_SLOPE * lg;
        atomicMaxF(&seg_max[r * NH + h], lg);
    }
}

// ---------------------------------------------------------------------------
// Kernel 4: fused exp + scatter. One wave per edge. Lane j owns features
// 4j..4j+3 (single head h = j>>3). Accumulates unnormalized weighted sum into
// out and the per-(node,head) denominator; normalization deferred to kernel 5.
// ---------------------------------------------------------------------------
__global__ void gat_edge_scatter(const int* __restrict__ snd,
                                 const int* __restrict__ rcv,
                                 const float* __restrict__ s_src,
                                 const float* __restrict__ s_dst,
                                 const float* __restrict__ seg_max,
                                 const float* __restrict__ X,
                                 float* __restrict__ denom,
                                 float* __restrict__ out,
                                 int E, int EP) {
    const int ew = blockIdx.x * (blockDim.x >> 5) + (threadIdx.x >> 5);
    if (ew >= EP) return;                       // uniform per wave
    const int lane = threadIdx.x & 31;
    const int s = (ew < E) ? snd[ew] : (ew - E);
    const int r = (ew < E) ? rcv[ew] : (ew - E);

    const int h = lane >> 3;                    // head for features 4*lane..+3
    float lg = s_src[s * NH + h] + s_dst[r * NH + h];
    lg = (lg > 0.0f) ? lg : NEG_SLOPE * lg;
    const float w = __expf(lg - seg_max[r * NH + h]);

    if ((lane & 7) == 0) atomicAdd(&denom[r * NH + h], w);

    const float4 sv = *reinterpret_cast<const float4*>(X + (size_t)s * HC + 4 * lane);
    float* op = out + (size_t)r * HC + 4 * lane;
    atomicAdd(op + 0, w * sv.x);
    atomicAdd(op + 1, w * sv.y);
    atomicAdd(op + 2, w * sv.z);
    atomicAdd(op + 3, w * sv.w);
}

// ---------------------------------------------------------------------------
// Kernel 5: out = out / denom[n,h] + bias[f]
// ---------------------------------------------------------------------------
__global__ void gat_finalize(float* __restrict__ out,
                             const float* __restrict__ denom,
                             const float* __restrict__ bias, int total) {
    const int idx = blockIdx.x * blockDim.x + threadIdx.x;
    if (idx >= total) return;
    const int n = idx >> 7;          // /128
    const int f = idx & 127;
    const int h = f >> 5;
    out[idx] = out[idx] / denom[n * NH + h] + bias[f];
}

extern "C" void kernel_launch(void* const* d_in, const int* in_sizes, int n_in,
                              void* d_out, int out_size, void* d_ws, size_t ws_size,
                              hipStream_t stream) {
    const float* nodes   = (const float*)d_in[0];
    const int*   senders = (const int*)d_in[1];
    const int*   recvers = (const int*)d_in[2];
    const float* Wm      = (const float*)d_in[3];
    const float* b_lin   = (const float*)d_in[4];
    const float* alpha   = (const float*)d_in[5];
    const float* bias    = (const float*)d_in[6];
    float* out = (float*)d_out;

    const int N  = in_sizes[0] / F_IN;   // 50000
    const int E  = in_sizes[1];          // 800000
    const int EP = E + N;                // with self loops

    // workspace layout (floats)
    float* ws      = (float*)d_ws;
    float* X       = ws;                              // N*128
    float* s_src   = X + (size_t)N * HC;              // N*4
    float* s_dst   = s_src + (size_t)N * NH;          // N*4
    float* seg_max = s_dst + (size_t)N * NH;          // N*4
    float* denom   = seg_max + (size_t)N * NH;        // N*4

    // 1) GEMM via WMMA: grid over 16-row tiles, 8 waves cover the 128 cols
    gat_gemm_wmma<<<(N + 15) / 16, 256, 0, stream>>>(nodes, Wm, b_lin, X, N);

    // 2) node scores + init
    gat_node_scores<<<N, 128, 0, stream>>>(X, alpha, s_src, s_dst,
                                           seg_max, denom, out, N);

    // 3) segment max over receivers
    gat_edge_max<<<(EP + 255) / 256, 256, 0, stream>>>(senders, recvers,
                                                       s_src, s_dst, seg_max, E, EP);

    // 4) fused exp + weighted scatter (one wave per edge, 8 edges per block)
    gat_edge_scatter<<<(EP + 7) / 8, 256, 0, stream>>>(senders, recvers,
                                                       s_src, s_dst, seg_max, X,
                                                       denom, out, E, EP);

    // 5) normalize + bias
    const int total = N * HC;
    gat_finalize<<<(total + 255) / 256, 256, 0, stream>>>(out, denom, bias, total);
}